// RadialFieldLayer_12472585027714
// MI455X (gfx1250) — compile-verified
//
#include <hip/hip_runtime.h>
#include <hip/hip_bf16.h>

typedef __attribute__((ext_vector_type(2))) float v2f;
typedef __attribute__((ext_vector_type(8))) float v8f;

__device__ __forceinline__ float lrelu02(float z) {
    return fmaxf(z, 0.0f) + 0.2f * fminf(z, 0.0f);
}

// ---------------- zero workspace ----------------
__global__ void rf_zero_kernel(float* __restrict__ p, int n) {
    int i = blockIdx.x * blockDim.x + threadIdx.x;
    if (i < n) p[i] = 0.0f;
}

// ---------------- edge kernel: WMMA f32 16x16x4 MLP + scatter atomics ----------------
__global__ __launch_bounds__(256) void rf_edge_kernel(
    const float* __restrict__ x, const int* __restrict__ ei,
    const float* __restrict__ Wp1, const float* __restrict__ bp1,
    const float* __restrict__ Wp2,
    float* __restrict__ agg, float* __restrict__ cnt, int E)
{
    const int tid = blockIdx.x * blockDim.x + threadIdx.x;
    const int lid = threadIdx.x & 31;
    const bool lo = (lid < 16);
    const int hi8 = lo ? 0 : 8;

    // Loop-invariant weight registers.
    // A_g (16x4 f32 A-matrix): row j' = Wp1[g*16+j'] in K-slot0, bp1 in K-slot1.
    v2f A[4];
    float w2[4][8];
#pragma unroll
    for (int g = 0; g < 4; ++g) {
        A[g].x = lo ? Wp1[g * 16 + lid] : 0.0f;
        A[g].y = lo ? bp1[g * 16 + lid] : 0.0f;
#pragma unroll
        for (int v = 0; v < 8; ++v)
            w2[g][v] = Wp2[g * 16 + v + hi8];
    }

    const bool act = tid < E;
    const int ec = act ? tid : 0;
    const int r0 = ei[ec];
    const int c0 = ei[E + ec];

    const float dx = x[3 * r0 + 0] - x[3 * c0 + 0];
    const float dy = x[3 * r0 + 1] - x[3 * c0 + 1];
    const float dz = x[3 * r0 + 2] - x[3 * c0 + 2];
    const float rad = sqrtf(dx * dx + dy * dy + dz * dz);
    const float rad_sw = __shfl_xor(rad, 16, 32);   // radii of the other half-wave's edges

    // B tiles (4x16 f32 B-matrix): row K=0 = radial of 16 edges, row K=1 = 1.0 (bias slot).
    v2f B0, B1;
    B0.x = lo ? rad : 0.0f;
    B0.y = lo ? 1.0f : 0.0f;
    B1.x = lo ? rad_sw : 0.0f;
    B1.y = B0.y;

    // D_g[j', e] = r_e*Wp1_j + bp1_j ; lrelu pointwise ; dot with per-lane Wp2 slices.
    float acc0 = 0.0f, acc1 = 0.0f;
#pragma unroll
    for (int g = 0; g < 4; ++g) {
        v8f cz = {};
        v8f d0 = __builtin_amdgcn_wmma_f32_16x16x4_f32(
            false, A[g], false, B0, (short)0, cz, false, false);
        v8f d1 = __builtin_amdgcn_wmma_f32_16x16x4_f32(
            false, A[g], false, B1, (short)0, cz, false, false);
#pragma unroll
        for (int v = 0; v < 8; ++v) {
            acc0 += lrelu02(d0[v]) * w2[g][v];
            acc1 += lrelu02(d1[v]) * w2[g][v];
        }
    }
    // Each edge's 64-term dot is split across lane pair (l, l^16): combine.
    const float y0 = acc0 + __shfl_xor(acc0, 16, 32);  // tile0: edges wavebase+0..15 (valid on lanes 0-15)
    const float y1 = acc1 + __shfl_xor(acc1, 16, 32);  // tile1: edges wavebase+16..31 (valid on lanes 16-31)
    const float eo = tanhf(lo ? y0 : y1);

    if (act) {
        atomicAdd(&agg[3 * r0 + 0], dx * eo);
        atomicAdd(&agg[3 * r0 + 1], dy * eo);
        atomicAdd(&agg[3 * r0 + 2], dz * eo);
        atomicAdd(&cnt[r0], 1.0f);
    }
}

// ---------------- node kernel: vel MLP + combine ----------------
__global__ __launch_bounds__(256) void rf_node_kernel(
    const float* __restrict__ x, const float* __restrict__ vel_norm,
    const float* __restrict__ vel,
    const float* __restrict__ W1, const float* __restrict__ b1,
    const float* __restrict__ W2, const float* __restrict__ b2,
    const float* __restrict__ agg, const float* __restrict__ cnt,
    float* __restrict__ out, int N)
{
    int i = blockIdx.x * blockDim.x + threadIdx.x;
    if (i >= N) return;
    const float vn = vel_norm[i];
    float s = b2[0];
#pragma unroll 8
    for (int j = 0; j < 64; ++j)
        s += W2[j] * lrelu02(fmaf(vn, W1[j], b1[j]));
    const float c = fmaxf(cnt[i], 1.0f);
#pragma unroll
    for (int k = 0; k < 3; ++k)
        out[3 * i + k] = (x[3 * i + k] + agg[3 * i + k] / c) + vel[3 * i + k] * s;
}

extern "C" void kernel_launch(void* const* d_in, const int* in_sizes, int n_in,
                              void* d_out, int out_size, void* d_ws, size_t ws_size,
                              hipStream_t stream) {
    const float* x        = (const float*)d_in[0];
    const float* vel_norm = (const float*)d_in[1];
    const float* vel      = (const float*)d_in[2];
    const int*   ei       = (const int*)d_in[3];
    const float* W1  = (const float*)d_in[4];
    const float* b1  = (const float*)d_in[5];
    const float* W2  = (const float*)d_in[6];
    const float* b2  = (const float*)d_in[7];
    const float* Wp1 = (const float*)d_in[8];
    const float* bp1 = (const float*)d_in[9];
    const float* Wp2 = (const float*)d_in[10];
    float* out = (float*)d_out;

    const int N = in_sizes[0] / 3;
    const int E = in_sizes[3] / 2;

    float* agg = (float*)d_ws;          // [N*3]
    float* cnt = agg + (size_t)3 * N;   // [N]

    const int zn = 4 * N;
    rf_zero_kernel<<<(zn + 255) / 256, 256, 0, stream>>>(agg, zn);
    rf_edge_kernel<<<(E + 255) / 256, 256, 0, stream>>>(
        x, ei, Wp1, bp1, Wp2, agg, cnt, E);
    rf_node_kernel<<<(N + 255) / 256, 256, 0, stream>>>(
        x, vel_norm, vel, W1, b1, W2, b2, agg, cnt, out, N);
}